// lbfgs_nn_31026843746379
// MI455X (gfx1250) — compile-verified
//
#include <hip/hip_runtime.h>

typedef __attribute__((ext_vector_type(2))) float v2f;
typedef __attribute__((ext_vector_type(8))) float v8f;

#define DDIM 4096
#define RANK 1024
#define NELEM ((size_t)DDIM * (size_t)DDIM)   // 16,777,216

// ---------------------------------------------------------------------------
// rho = U U^H  (unnormalized; 1/trace folded into final gather)
//   rho_r = Ur Ur^T + Ui Ui^T      (symmetric)
//   rho_i = Ui Ur^T - Ur Ui^T      (antisymmetric)
// rho is Hermitian -> compute only block tiles with m-tile >= n-tile (2080 of
// 4096) and emit the conjugate-transposed mirror tile via an LDS transpose
// (coalesced stores). WMMA f32 16x16x4; wave tile 32x32; block = 4 waves =
// 64x64 so the triangular cut is symmetric.
// ---------------------------------------------------------------------------
__global__ __launch_bounds__(128)
void rho_gemm_wmma_herm(const float* __restrict__ P,
                        float* __restrict__ rho_r, float* __restrict__ rho_i) {
  if (blockIdx.x > blockIdx.y) return;   // block-uniform: keep m-tile >= n-tile

  const float* Ur = P;
  const float* Ui = P + (size_t)DDIM * RANK;

  const int lane = threadIdx.x & 31;
  const int wave = threadIdx.x >> 5;
  const int l    = lane & 15;   // M/N index within fragment
  const int half = lane >> 4;   // K-pair select (K0,K1 | K2,K3)

  const int m0 = blockIdx.y * 64 + (wave >> 1) * 32;
  const int n0 = blockIdx.x * 64 + (wave & 1) * 32;

  const v8f z = {0.f,0.f,0.f,0.f,0.f,0.f,0.f,0.f};
  v8f cr[2][2] = {{z,z},{z,z}};
  v8f ci[2][2] = {{z,z},{z,z}};

  for (int kb = 0; kb < RANK; kb += 4) {
    const int koff = kb + half * 2;

    v2f ar[2], ai[2], br[2], bi[2], nbi[2];
#pragma unroll
    for (int mi = 0; mi < 2; ++mi) {
      const size_t ra = (size_t)(m0 + mi * 16 + l) * RANK + koff;
      ar[mi] = *(const v2f*)(Ur + ra);
      ai[mi] = *(const v2f*)(Ui + ra);
      __builtin_prefetch(Ur + ra + 8, 0, 1);   // global_prefetch_b8
    }
#pragma unroll
    for (int ni = 0; ni < 2; ++ni) {
      const size_t rb = (size_t)(n0 + ni * 16 + l) * RANK + koff;
      br[ni]  = *(const v2f*)(Ur + rb);
      bi[ni]  = *(const v2f*)(Ui + rb);
      nbi[ni] = -bi[ni];                       // fp32 WMMA has no A/B neg bits
    }
#pragma unroll
    for (int mi = 0; mi < 2; ++mi) {
#pragma unroll
      for (int ni = 0; ni < 2; ++ni) {
        cr[mi][ni] = __builtin_amdgcn_wmma_f32_16x16x4_f32(
            false, ar[mi], false, br[ni], (short)0, cr[mi][ni], false, false);
        cr[mi][ni] = __builtin_amdgcn_wmma_f32_16x16x4_f32(
            false, ai[mi], false, bi[ni], (short)0, cr[mi][ni], false, false);
        ci[mi][ni] = __builtin_amdgcn_wmma_f32_16x16x4_f32(
            false, ai[mi], false, br[ni], (short)0, ci[mi][ni], false, false);
        ci[mi][ni] = __builtin_amdgcn_wmma_f32_16x16x4_f32(
            false, ar[mi], false, nbi[ni], (short)0, ci[mi][ni], false, false);
      }
    }
  }

  // ---- direct store of the (m,n) tile -------------------------------------
  // C/D layout: VGPR v, lanes 0-15 -> M=v, N=lane; lanes 16-31 -> M=v+8
#pragma unroll
  for (int mi = 0; mi < 2; ++mi)
#pragma unroll
    for (int ni = 0; ni < 2; ++ni)
#pragma unroll
      for (int v = 0; v < 8; ++v) {
        const size_t row = (size_t)(m0 + mi * 16 + half * 8 + v);
        const size_t col = (size_t)(n0 + ni * 16 + l);
        rho_r[row * DDIM + col] = cr[mi][ni][v];
        rho_i[row * DDIM + col] = ci[mi][ni][v];
      }

  // ---- mirror tile (n,m) = conj: transpose through LDS for coalescing -----
  if (blockIdx.x != blockIdx.y) {            // block-uniform
    __shared__ float t[4][32][33];           // +1 pad: no LDS bank conflicts
    // real plane
#pragma unroll
    for (int mi = 0; mi < 2; ++mi)
#pragma unroll
      for (int ni = 0; ni < 2; ++ni)
#pragma unroll
        for (int v = 0; v < 8; ++v)
          t[wave][mi * 16 + half * 8 + v][ni * 16 + l] = cr[mi][ni][v];
    __syncthreads();
#pragma unroll
    for (int mi = 0; mi < 2; ++mi)
#pragma unroll
      for (int ni = 0; ni < 2; ++ni)
#pragma unroll
        for (int v = 0; v < 8; ++v) {
          const int rT = mi * 16 + half * 8 + v;   // local row in mirror tile
          const int cT = ni * 16 + l;              // local col (lane-varying)
          rho_r[(size_t)(n0 + rT) * DDIM + (size_t)(m0 + cT)] = t[wave][cT][rT];
        }
    __syncthreads();
    // imag plane (negated: conjugate)
#pragma unroll
    for (int mi = 0; mi < 2; ++mi)
#pragma unroll
      for (int ni = 0; ni < 2; ++ni)
#pragma unroll
        for (int v = 0; v < 8; ++v)
          t[wave][mi * 16 + half * 8 + v][ni * 16 + l] = -ci[mi][ni][v];
    __syncthreads();
#pragma unroll
    for (int mi = 0; mi < 2; ++mi)
#pragma unroll
      for (int ni = 0; ni < 2; ++ni)
#pragma unroll
        for (int v = 0; v < 8; ++v) {
          const int rT = mi * 16 + half * 8 + v;
          const int cT = ni * 16 + l;
          rho_i[(size_t)(n0 + rT) * DDIM + (size_t)(m0 + cT)] = t[wave][cT][rT];
        }
  }
}

// ---------------------------------------------------------------------------
// trace(U U^H) = sum(params^2) -- deterministic two-stage reduction
// ---------------------------------------------------------------------------
__global__ __launch_bounds__(256)
void trace_partial(const float* __restrict__ P, float* __restrict__ part, size_t n) {
  __shared__ float sm[256];
  float acc = 0.f;
  for (size_t i = (size_t)blockIdx.x * 256 + threadIdx.x; i < n;
       i += (size_t)gridDim.x * 256) {
    const float x = P[i];
    acc += x * x;
  }
  sm[threadIdx.x] = acc;
  __syncthreads();
  for (int off = 128; off > 0; off >>= 1) {
    if ((int)threadIdx.x < off) sm[threadIdx.x] += sm[threadIdx.x + off];
    __syncthreads();
  }
  if (threadIdx.x == 0) part[blockIdx.x] = sm[0];
}

__global__ __launch_bounds__(256)
void trace_final(const float* __restrict__ part, float* __restrict__ tr, int n) {
  __shared__ float sm[256];
  float acc = 0.f;
  for (int i = threadIdx.x; i < n; i += 256) acc += part[i];
  sm[threadIdx.x] = acc;
  __syncthreads();
  for (int off = 128; off > 0; off >>= 1) {
    if ((int)threadIdx.x < off) sm[threadIdx.x] += sm[threadIdx.x + off];
    __syncthreads();
  }
  if (threadIdx.x == 0) tr[0] = sm[0];
}

// ---------------------------------------------------------------------------
// Fused 2-qubit contraction pass, WMMA formulation.
//   out[16 s-combos, site] = W(16x16 cpx) x gathered(16 in, site)
// where W[(s1,s2),(j1,i1,j2,i2)] = M[s1,i1,j1]*M[s2,i2,j2]  (constant).
// Each wave owns 16 sites (one 16x16x16 complex GEMM = 16 WMMA f32 16x16x4).
// ---------------------------------------------------------------------------
__global__ __launch_bounds__(256)
void qmt_pass2_wmma(const float* __restrict__ Mr, const float* __restrict__ Mi,
                    const float* __restrict__ sr, const float* __restrict__ si,
                    float* __restrict__ dr, float* __restrict__ di,
                    int R1, int R2, int C1, int C2, int C_in,
                    int spk_shift, int cout_shift, int store_imag) {
  const int lane = threadIdx.x & 31;
  const int half = lane >> 4;       // selects K-pair {0,1} vs {2,3} in frags
  const int lm   = lane & 15;       // M (=s) for A-frag, N (=site) for B/C

  // ---- A fragments: W real, W imag, -W imag (3 x 4 chunks x v2f) ----
  v2f aWr[4], aWi[4], aWn[4];
  {
    const int s1 = lm >> 2, s2 = lm & 3;
#pragma unroll
    for (int c = 0; c < 4; ++c) {
      v2f er, ei;
#pragma unroll
      for (int e = 0; e < 2; ++e) {
        const int in = c * 4 + half * 2 + e;
        const int j1 = (in >> 3) & 1, i1 = (in >> 2) & 1;
        const int j2 = (in >> 1) & 1, i2 = in & 1;
        const float m1r = Mr[s1 * 4 + i1 * 2 + j1], m1i = Mi[s1 * 4 + i1 * 2 + j1];
        const float m2r = Mr[s2 * 4 + i2 * 2 + j2], m2i = Mi[s2 * 4 + i2 * 2 + j2];
        er[e] = m1r * m2r - m1i * m2i;
        ei[e] = m1r * m2i + m1i * m2r;
      }
      aWr[c] = er;
      aWi[c] = ei;
      aWn[c] = -ei;
    }
  }

  // ---- per-lane site / address decode (shifts only; dims are powers of 4) --
  const size_t tile    = (size_t)blockIdx.x * 8 + (threadIdx.x >> 5);
  const size_t siteIdx = tile * 16 + lm;               // 1,048,576 sites total
  const size_t k   = siteIdx >> spk_shift;
  const int    rem = (int)(siteIdx & (((size_t)1 << spk_shift) - 1));
  const int    a2  = rem >> cout_shift;
  const int    t2  = rem & ((1 << cout_shift) - 1);
  const int    R_in = R1 * 2;
  const size_t base0 = (k * (size_t)R_in + (size_t)a2) * (size_t)C_in + (size_t)t2;

  const v8f z = {0.f,0.f,0.f,0.f,0.f,0.f,0.f,0.f};
  v8f outr = z, outi = z;

#pragma unroll
  for (int c = 0; c < 4; ++c) {
    v2f bvr, bvi;
#pragma unroll
    for (int e = 0; e < 2; ++e) {
      const int in = c * 4 + half * 2 + e;
      const int j1 = (in >> 3) & 1, i1 = (in >> 2) & 1;
      const int j2 = (in >> 1) & 1, i2 = in & 1;
      const size_t off = base0 +
          (size_t)((j1 * R1 + j2 * R2) * C_in + i1 * C1 + i2 * C2);
      bvr[e] = sr[off];
      bvi[e] = si[off];
    }
    // out_r += Wr*vr - Wi*vi ; out_i += Wr*vi + Wi*vr
    outr = __builtin_amdgcn_wmma_f32_16x16x4_f32(
        false, aWr[c], false, bvr, (short)0, outr, false, false);
    outr = __builtin_amdgcn_wmma_f32_16x16x4_f32(
        false, aWn[c], false, bvi, (short)0, outr, false, false);
    outi = __builtin_amdgcn_wmma_f32_16x16x4_f32(
        false, aWr[c], false, bvi, (short)0, outi, false, false);
    outi = __builtin_amdgcn_wmma_f32_16x16x4_f32(
        false, aWi[c], false, bvr, (short)0, outi, false, false);
  }

  // ---- store: D layout lane holds site=lm, s = v + half*8 ------------------
  const size_t spk   = (size_t)1 << spk_shift;         // R_out*C_out
  const size_t obase = (k << (spk_shift + 4)) + (size_t)rem;
#pragma unroll
  for (int v = 0; v < 8; ++v) {
    const size_t o = obase + (size_t)(v + half * 8) * spk;
    dr[o] = outr[v];
    if (store_imag) di[o] = outi[v];
  }
}

// ---------------------------------------------------------------------------
// out[i] = T_real[idx[i]] / trace
// ---------------------------------------------------------------------------
__global__ __launch_bounds__(256)
void gather_scale(const float* __restrict__ Tre, const int* __restrict__ idx,
                  const float* __restrict__ tr, float* __restrict__ out, int n) {
  const int i = blockIdx.x * 256 + threadIdx.x;
  if (i < n) out[i] = Tre[idx[i]] * (1.0f / tr[0]);
}

// ---------------------------------------------------------------------------
extern "C" void kernel_launch(void* const* d_in, const int* in_sizes, int n_in,
                              void* d_out, int out_size, void* d_ws, size_t ws_size,
                              hipStream_t stream) {
  (void)n_in; (void)ws_size; (void)out_size;
  const float* params = (const float*)d_in[0];   // (2, 4096, 1024)
  const float* Mr     = (const float*)d_in[1];   // (4,2,2)
  const float* Mi     = (const float*)d_in[2];   // (4,2,2)
  const int*   idx    = (const int*)d_in[3];     // (1e6,)
  float* out = (float*)d_out;
  const int n_idx = in_sizes[3];

  float* buf0r = (float*)d_ws;          // rho_r then ping-pong
  float* buf0i = buf0r + NELEM;
  float* buf1r = buf0i + NELEM;
  float* buf1i = buf1r + NELEM;
  float* part  = buf1i + NELEM;         // 1024 floats
  float* trace = part + 1024;           // 1 float

  // trace
  trace_partial<<<1024, 256, 0, stream>>>(params, part, (size_t)2 * DDIM * RANK);
  trace_final<<<1, 256, 0, stream>>>(part, trace, 1024);

  // rho = U U^H (unnormalized, Hermitian) into buf0: lower-triangle blocks
  // computed, upper half mirrored through LDS transpose.
  rho_gemm_wmma_herm<<<dim3(DDIM / 64, DDIM / 64), 128, 0, stream>>>(
      params, buf0r, buf0i);

  // 6 fused 2-qubit passes, ping-pong buf0 <-> buf1; pass 5 lands in buf0.
  // 65,536 wave-tiles per pass / 8 waves per block = 8192 blocks.
  for (int p = 0; p < 6; ++p) {
    const int R_in = DDIM >> (2 * p);
    const int R1 = R_in >> 1, R2 = R_in >> 2;
    const int C1 = R1, C2 = R2;
    const int C_in = R_in;
    const int spk_shift  = 20 - 4 * p;   // log2(R_out*C_out)
    const int cout_shift = 10 - 2 * p;   // log2(C_out)
    const int store_imag = (p < 5) ? 1 : 0;  // final pass: real part only
    const float* sr = (p & 1) ? buf1r : buf0r;
    const float* si = (p & 1) ? buf1i : buf0i;
    float* dr = (p & 1) ? buf0r : buf1r;
    float* di = (p & 1) ? buf0i : buf1i;
    qmt_pass2_wmma<<<8192, 256, 0, stream>>>(Mr, Mi, sr, si, dr, di,
                                             R1, R2, C1, C2, C_in,
                                             spk_shift, cout_shift, store_imag);
  }

  // gather + normalize (P is linear in rho, so dividing at the end is exact)
  gather_scale<<<(n_idx + 255) / 256, 256, 0, stream>>>(buf0r, idx, trace, out, n_idx);
}